// VariationalMeanField_42932493090864
// MI455X (gfx1250) — compile-verified
//
#include <hip/hip_runtime.h>
#include <math.h>

#define BATCH   1048576
#define NSAMPLE 30

typedef __attribute__((ext_vector_type(2))) float v2f;
typedef __attribute__((ext_vector_type(8))) float v8f;

#if defined(__gfx1250__)
#if __has_builtin(__builtin_amdgcn_wmma_f32_16x16x4_f32)
#define USE_WMMA_F32 1
#endif
#define USE_ASYNC_LDS 1
#endif

// ---- LDS layout (floats) ----
#define O_MW0   0
#define O_MB0   40
#define O_MW00  80
#define O_MB00  880
#define O_MW1   900
#define O_MB1   940
#define O_SW0   942
#define O_SB0   982
#define O_SW00  1022
#define O_SB00  1822
#define O_SW1   1842
#define O_SB1   1882
#define LDS_FLOATS 1884

__device__ __forceinline__ float softplus_f(float h) {
  return (h > 20.f) ? h : log1pf(__expf(h));
}

// =====================================================================
// Kernel 1: fused 2-net MLP (1->40->20->2 + softplus) via fp32 WMMA.
// Each wave computes a 16-row tile; layer-2 GEMM (K=40, N=20) is
// 10 K-steps x 2 N-tiles x 2 nets of V_WMMA_F32_16X16X4_F32.
// w00 weight staging uses gfx1250 async global->LDS DMA (ASYNCcnt).
// =====================================================================
__global__ void __launch_bounds__(256) mlp_wmma_kernel(
    const float* __restrict__ state,
    const float* __restrict__ mw0,  const float* __restrict__ mb0,
    const float* __restrict__ mw00, const float* __restrict__ mb00,
    const float* __restrict__ mw1,  const float* __restrict__ mb1,
    const float* __restrict__ sw0,  const float* __restrict__ sb0,
    const float* __restrict__ sw00, const float* __restrict__ sb00,
    const float* __restrict__ sw1,  const float* __restrict__ sb1,
    float* __restrict__ xm, float* __restrict__ xs, double* __restrict__ acc)
{
  __shared__ float L[LDS_FLOATS];
  for (int i = threadIdx.x; i < 40; i += 256) {
    L[O_MW0+i]=mw0[i]; L[O_MB0+i]=mb0[i]; L[O_SW0+i]=sw0[i]; L[O_SB0+i]=sb0[i];
    L[O_MW1+i]=mw1[i]; L[O_SW1+i]=sw1[i];
  }
#ifdef USE_ASYNC_LDS
  {
    // Async global->LDS copy of the two 40x20 weight matrices (ASYNCcnt path).
    // VGPR operand 0 = LDS byte address (low 32 bits of the AS3 pointer),
    // VGPR operand 1 = 32-bit byte offset, SGPR pair = base address (GVS mode).
    const unsigned lm  = (unsigned)(unsigned long long)(&L[O_MW00]);
    const unsigned lsg = (unsigned)(unsigned long long)(&L[O_SW00]);
    for (int i = threadIdx.x; i < 800; i += 256) {
      const unsigned boff = (unsigned)(i * 4);
      asm volatile("global_load_async_to_lds_b32 %0, %1, %2 offset:0"
                   :: "v"(lm + boff), "v"(boff), "s"(mw00) : "memory");
      asm volatile("global_load_async_to_lds_b32 %0, %1, %2 offset:0"
                   :: "v"(lsg + boff), "v"(boff), "s"(sw00) : "memory");
    }
    asm volatile("s_wait_asynccnt 0x0" ::: "memory");
  }
#else
  for (int i = threadIdx.x; i < 800; i += 256) { L[O_MW00+i]=mw00[i]; L[O_SW00+i]=sw00[i]; }
#endif
  if (threadIdx.x < 20) { L[O_MB00+threadIdx.x]=mb00[threadIdx.x]; L[O_SB00+threadIdx.x]=sb00[threadIdx.x]; }
  if (threadIdx.x < 2)  { L[O_MB1+threadIdx.x]=mb1[threadIdx.x];  L[O_SB1+threadIdx.x]=sb1[threadIdx.x]; }
  if (blockIdx.x == 0 && threadIdx.x == 0) *acc = 0.0;   // reset score accumulator
  __syncthreads();

#ifdef USE_WMMA_F32
  const int lane = threadIdx.x & 31;
  const int wave = threadIdx.x >> 5;
  const int m    = lane & 15;     // M index (row within tile) / N index (col)
  const int half = lane >> 4;     // K-half of the fragment
  const int mh   = 16 + (m & 3);  // always-valid N-tile-1 column (16..19)
  const float hiMask = (m < 4) ? 1.f : 0.f;  // zero out lanes m>=4 in tile 1
  const long long row0 = ((long long)blockIdx.x * 8 + wave) * 16;
  const float s = state[row0 + m];

  v8f cm0 = {}, cm1 = {}, cs0 = {}, cs1 = {};
#pragma unroll
  for (int k = 0; k < 10; ++k) {
    const int kb = 4*k + 2*half;          // A lanes 0-15: K=4k,4k+1; lanes 16-31: K=4k+2,4k+3
    v2f am, az, bm0, bm1, bs0, bs1;
    am[0] = fmaf(s, L[O_MW0+kb],   L[O_MB0+kb]);     // layer-1 generated in-register
    am[1] = fmaf(s, L[O_MW0+kb+1], L[O_MB0+kb+1]);
    az[0] = fmaf(s, L[O_SW0+kb],   L[O_SB0+kb]);
    az[1] = fmaf(s, L[O_SW0+kb+1], L[O_SB0+kb+1]);
    bm0[0] = L[O_MW00 + kb*20 + m];                  // B 4x16: row K, col n=m
    bm0[1] = L[O_MW00 + (kb+1)*20 + m];
    bs0[0] = L[O_SW00 + kb*20 + m];
    bs0[1] = L[O_SW00 + (kb+1)*20 + m];
    bm1[0] = L[O_MW00 + kb*20 + mh]     * hiMask;    // unconditional load + mask (no exec churn)
    bm1[1] = L[O_MW00 + (kb+1)*20 + mh] * hiMask;
    bs1[0] = L[O_SW00 + kb*20 + mh]     * hiMask;
    bs1[1] = L[O_SW00 + (kb+1)*20 + mh] * hiMask;

    cm0 = __builtin_amdgcn_wmma_f32_16x16x4_f32(false, am, false, bm0, (short)0, cm0, false, false);
    cm1 = __builtin_amdgcn_wmma_f32_16x16x4_f32(false, am, false, bm1, (short)0, cm1, false, false);
    cs0 = __builtin_amdgcn_wmma_f32_16x16x4_f32(false, az, false, bs0, (short)0, cs0, false, false);
    cs1 = __builtin_amdgcn_wmma_f32_16x16x4_f32(false, az, false, bs1, (short)0, cs1, false, false);
  }

  // Layer 3 (20 -> 2): per-lane column n = m (tile0) / 16+m (tile1), reduce across 16 lanes.
  const float wmA  = L[O_MW1 + m*2 + 0], wmB = L[O_MW1 + m*2 + 1];
  const float wsA  = L[O_SW1 + m*2 + 0], wsB = L[O_SW1 + m*2 + 1];
  const float wmAh = L[O_MW1 + mh*2 + 0] * hiMask;
  const float wmBh = L[O_MW1 + mh*2 + 1] * hiMask;
  const float wsAh = L[O_SW1 + mh*2 + 0] * hiMask;
  const float wsBh = L[O_SW1 + mh*2 + 1] * hiMask;
  const float bmc  = L[O_MB00 + m], bmch = L[O_MB00 + mh] * hiMask;
  const float bsc  = L[O_SB00 + m], bsch = L[O_SB00 + mh] * hiMask;
  const float bm1a = L[O_MB1+0], bm1b = L[O_MB1+1];
  const float bs1a = L[O_SB1+0], bs1b = L[O_SB1+1];

#pragma unroll
  for (int r = 0; r < 8; ++r) {
    float hm0 = cm0[r] + bmc,  hm1 = cm1[r] + bmch;
    float hs0 = cs0[r] + bsc,  hs1 = cs1[r] + bsch;
    float pm0 = hm0*wmA + hm1*wmAh;
    float pm1 = hm0*wmB + hm1*wmBh;
    float ps0 = hs0*wsA + hs1*wsAh;
    float ps1 = hs0*wsB + hs1*wsBh;
#pragma unroll
    for (int msk = 1; msk < 16; msk <<= 1) {   // reduce within each 16-lane half
      pm0 += __shfl_xor(pm0, msk);
      pm1 += __shfl_xor(pm1, msk);
      ps0 += __shfl_xor(ps0, msk);
      ps1 += __shfl_xor(ps1, msk);
    }
    if (m == r) {
      const long long row = row0 + half*8 + r;   // C rows: lanes<16 -> M=r, lanes>=16 -> M=8+r
      xm[row*2+0] = softplus_f(pm0 + bm1a);
      xm[row*2+1] = softplus_f(pm1 + bm1b);
      xs[row*2+0] = softplus_f(ps0 + bs1a);
      xs[row*2+1] = softplus_f(ps1 + bs1b);
    }
  }
#else
  // Scalar fallback (no fp32 WMMA builtin available)
  const long long gid = (long long)blockIdx.x * blockDim.x + threadIdx.x;
  if (gid < (long long)BATCH) {
    const float s = state[gid];
    float h1m[40], h1s[40];
    for (int j = 0; j < 40; ++j) {
      h1m[j] = fmaf(s, L[O_MW0+j], L[O_MB0+j]);
      h1s[j] = fmaf(s, L[O_SW0+j], L[O_SB0+j]);
    }
    float om0=0.f, om1=0.f, os0=0.f, os1=0.f;
    for (int n = 0; n < 20; ++n) {
      float hm = L[O_MB00+n], hs = L[O_SB00+n];
      for (int j = 0; j < 40; ++j) {
        hm = fmaf(h1m[j], L[O_MW00 + j*20 + n], hm);
        hs = fmaf(h1s[j], L[O_SW00 + j*20 + n], hs);
      }
      om0 = fmaf(hm, L[O_MW1 + n*2 + 0], om0);
      om1 = fmaf(hm, L[O_MW1 + n*2 + 1], om1);
      os0 = fmaf(hs, L[O_SW1 + n*2 + 0], os0);
      os1 = fmaf(hs, L[O_SW1 + n*2 + 1], os1);
    }
    xm[gid*2+0] = softplus_f(om0 + L[O_MB1+0]);
    xm[gid*2+1] = softplus_f(om1 + L[O_MB1+1]);
    xs[gid*2+0] = softplus_f(os0 + L[O_SB1+0]);
    xs[gid*2+1] = softplus_f(os1 + L[O_SB1+1]);
  }
#endif
}

// =====================================================================
// Kernel 2: per-row gamma sampling + log-prob score (VALU-trans bound).
// All transcendentals use hardware v_log_f32/v_exp_f32; lgamma is a
// branchless shift-ladder + 3-term Stirling series (2 logs, ~35 VALU).
// =====================================================================
__device__ __forceinline__ unsigned long long mix64(unsigned long long z) {
  z += 0x9E3779B97F4A7C15ull;
  z = (z ^ (z >> 30)) * 0xBF58476D1CE4E5B9ull;
  z = (z ^ (z >> 27)) * 0x94D049BB133111EBull;
  return z ^ (z >> 31);
}
__device__ __forceinline__ float u01f(unsigned int x) {
  return ((x >> 8) + 0.5f) * (1.0f / 16777216.0f);   // (0,1)
}

__device__ __forceinline__ float fast_lgammaf(float z) {
  // Branchless shift into Stirling region (z >= 8): lgamma(z) = lgamma(z+n) - log(prod)
  float p = 1.f, zz = z;
#pragma unroll
  for (int i = 0; i < 8; ++i) {
    const bool sh = (zz < 8.f);
    p  = sh ? p * zz : p;
    zz = sh ? zz + 1.f : zz;
  }
  const float inv  = __frcp_rn(zz);
  const float inv2 = inv * inv;
  const float ser  = inv * fmaf(inv2, fmaf(inv2, (1.f/1260.f), -(1.f/360.f)), (1.f/12.f));
  // (z-1/2)ln z - z + ln(2*pi)/2 + series - ln(shift product)
  return fmaf(zz - 0.5f, __logf(zz), -zz) + 0.91893853320467274f + ser - __logf(p);
}

__device__ float gamma_sample(float a, unsigned long long key) {
  const float aa = (a < 1.f) ? a + 1.f : a;
  const float d  = aa - (1.f / 3.f);
  const float c  = rsqrtf(9.f * d);
  float g = d;                       // deterministic fallback
#pragma unroll 1
  for (int t = 0; t < 8; ++t) {
    const unsigned long long r0 = mix64(key ^ (0xD1342543DE82EF95ull * (unsigned long long)(t + 1)));
    const unsigned long long r1 = mix64(r0 + 0x9E3779B97F4A7C15ull);
    const float u1 = u01f((unsigned int)r0);
    const float u2 = u01f((unsigned int)(r0 >> 32));
    const float u3 = u01f((unsigned int)r1);
    const float n  = sqrtf(-2.f * __logf(u1)) * __cosf(6.2831853071795865f * u2);
    float v = 1.f + c * n;
    if (v <= 0.f) continue;
    v = v * v * v;
    if (__logf(u3) < 0.5f * n * n + d * (1.f - v + __logf(v))) { g = d * v; break; }
  }
  if (a < 1.f) {
    const unsigned long long r2 = mix64(key + 0x5851F42D4C957F2Dull);
    g *= __expf(__logf(u01f((unsigned int)r2)) * __frcp_rn(a));  // u^(1/a) boost
  }
  return g;
}

__global__ void __launch_bounds__(256) score_kernel(
    const float* __restrict__ state, const float* __restrict__ xm,
    const float* __restrict__ xs, double* __restrict__ acc)
{
  const int gid = blockIdx.x * 256 + threadIdx.x;
  float sc = 0.f;
  if (gid < BATCH) {
    const float s   = state[gid];
    const float a_a = fmaxf(xm[2*gid+0], 1e-6f), r_a = fmaxf(xm[2*gid+1], 1e-6f);
    const float a_b = fmaxf(xs[2*gid+0], 1e-6f), r_b = fmaxf(xs[2*gid+1], 1e-6f);
    const float lgaa = fast_lgammaf(a_a), lgab = fast_lgammaf(a_b);
    const float lra  = __logf(r_a),       lrb  = __logf(r_b);
    const float ls   = __logf(s),         l1ms = log1pf(-s);
    float accf = 0.f;
#pragma unroll 1
    for (int i = 0; i < NSAMPLE; ++i) {
      const unsigned long long ka = mix64(((unsigned long long)gid << 6) | (unsigned)(2*i + 0));
      const unsigned long long kb = mix64(((unsigned long long)gid << 6) | (unsigned)(2*i + 1));
      const float za = fmaxf(gamma_sample(a_a, ka) / r_a, 1e-20f);
      const float zb = fmaxf(gamma_sample(a_b, kb) / r_b, 1e-20f);
      const float lza = __logf(za), lzb = __logf(zb);
      const float gamal = a_a*lra + (a_a - 1.f)*lza - r_a*za - lgaa;
      const float gambe = a_b*lrb + (a_b - 1.f)*lzb - r_b*zb - lgab;
      const float alp = 79.f*lza - 20.f*za - 29.632515766701894f;  // 80ln20 - lgamma(80)
      const float bet = lzb - 2.f*zb + 1.3862943611198906f;        // 2ln2 - lgamma(2)
      const float lbeta = fast_lgammaf(za) + fast_lgammaf(zb) - fast_lgammaf(za + zb);
      const float blp = (za - 1.f)*ls + (zb - 1.f)*l1ms - lbeta;
      accf += alp + bet - gamal - gambe + blp;
    }
    sc = accf * (1.f / (float)NSAMPLE);
  }
#pragma unroll
  for (int msk = 16; msk >= 1; msk >>= 1) sc += __shfl_xor(sc, msk);
  __shared__ float wsum[8];
  if ((threadIdx.x & 31) == 0) wsum[threadIdx.x >> 5] = sc;
  __syncthreads();
  if (threadIdx.x == 0) {
    float t = 0.f;
    for (int w = 0; w < 8; ++w) t += wsum[w];
    atomicAdd(acc, (double)t);
  }
}

__global__ void finalize_kernel(const double* __restrict__ acc, float* __restrict__ out) {
  out[0] = (float)(acc[0] / (double)BATCH);
}

extern "C" void kernel_launch(void* const* d_in, const int* in_sizes, int n_in,
                              void* d_out, int out_size, void* d_ws, size_t ws_size,
                              hipStream_t stream) {
  const float* state = (const float*)d_in[0];
  const float* mw0   = (const float*)d_in[1];
  const float* mb0   = (const float*)d_in[2];
  const float* mw00  = (const float*)d_in[3];
  const float* mb00  = (const float*)d_in[4];
  const float* mw1   = (const float*)d_in[5];
  const float* mb1   = (const float*)d_in[6];
  const float* sw0   = (const float*)d_in[7];
  const float* sb0   = (const float*)d_in[8];
  const float* sw00  = (const float*)d_in[9];
  const float* sb00  = (const float*)d_in[10];
  const float* sw1   = (const float*)d_in[11];
  const float* sb1   = (const float*)d_in[12];

  float* out = (float*)d_out;                 // [0]=score, then x_m (B,2), x_s (B,2)
  float* xm  = out + 1;
  float* xs  = out + 1 + 2 * (size_t)BATCH;
  double* acc = (double*)d_ws;

  mlp_wmma_kernel<<<BATCH / 128, 256, 0, stream>>>(state, mw0, mb0, mw00, mb00, mw1, mb1,
                                                   sw0, sb0, sw00, sb00, sw1, sb1,
                                                   xm, xs, acc);
  score_kernel<<<BATCH / 256, 256, 0, stream>>>(state, xm, xs, acc);
  finalize_kernel<<<1, 1, 0, stream>>>(acc, out);
}